// GRN_15307263443310
// MI455X (gfx1250) — compile-verified
//
#include <hip/hip_runtime.h>

typedef __attribute__((ext_vector_type(2))) float v2f;
typedef __attribute__((ext_vector_type(8))) float v8f;

#define D          128   // D_IN == D_OUT
#define TILE_M     16
#define THREADS    256   // 8 wave32 waves
#define LDS_STRIDE 132   // 128 + 4 pad: lanes land on distinct LDS banks

__device__ __forceinline__ int lower_bound_i32(const int* __restrict__ seg, int n, int target) {
    int lo = 0, hi = n;
    while (lo < hi) {
        int mid = (lo + hi) >> 1;
        if (seg[mid] < target) lo = mid + 1; else hi = mid;
    }
    return lo;
}

__global__ __launch_bounds__(THREADS)
void grn_fused_kernel(const float* __restrict__ feats,   // [E,128]
                      const float* __restrict__ att,     // [E]
                      const float* __restrict__ W,       // [128,128] row-major (k,n)
                      const float* __restrict__ bias,    // [128]
                      const int*   __restrict__ seg,     // [E], sorted
                      float*       __restrict__ out,     // [N,128]
                      int n_edges)
{
    __shared__ float As[TILE_M][LDS_STRIDE];

    const int lane = threadIdx.x & 31;
    const int wave = threadIdx.x >> 5;
    const int row0 = blockIdx.x * TILE_M;

    // ---------------- Phase 1: weighted segment-sum into LDS S-tile -----------
    // Wave w reduces nodes row0+2w and row0+2w+1. Lane l owns feats [4l, 4l+4).
    // Each 128-float edge row is one coalesced 512B wave load (global_load_b128).
    for (int t = 0; t < 2; ++t) {
        const int m    = wave * 2 + t;
        const int node = row0 + m;
        const int s = lower_bound_i32(seg, n_edges, node);
        const int e = lower_bound_i32(seg, n_edges, node + 1);
        float4 acc = make_float4(0.f, 0.f, 0.f, 0.f);
        for (int i = s; i < e; ++i) {
            const float  sc = att[i];
            const float4 xv = ((const float4*)(feats + (size_t)i * D))[lane];
            acc.x = fmaf(sc, xv.x, acc.x);
            acc.y = fmaf(sc, xv.y, acc.y);
            acc.z = fmaf(sc, xv.z, acc.z);
            acc.w = fmaf(sc, xv.w, acc.w);
        }
        *(float4*)&As[m][lane * 4] = acc;
    }
    __syncthreads();

    // ---------------- Phase 2: S_tile(16x128) @ W(128x128) via WMMA f32 -------
    // Wave w computes output columns [16w, 16w+16). K-loop: 32 x v_wmma_f32_16x16x4_f32.
    // A layout (16x4 f32): lane (half,idx): VGPR0/1 = A[idx][k0+2*half +0/+1]
    // B layout (4x16 f32): mirrored:        VGPR0/1 = B[k0+2*half +0/+1][idx]
    const int half = lane >> 4;   // 0: lanes 0-15, 1: lanes 16-31
    const int idx  = lane & 15;
    const int col  = wave * 16 + idx;

    v8f c = {};
    #pragma unroll
    for (int k0 = 0; k0 < D; k0 += 4) {
        const int k = k0 + 2 * half;
        v2f a, b;
        a.x = As[idx][k];
        a.y = As[idx][k + 1];
        b.x = W[(size_t)k * D + col];
        b.y = W[(size_t)(k + 1) * D + col];
        // 8 args: (neg_a, A, neg_b, B, c_mod, C, reuse_a, reuse_b)
        c = __builtin_amdgcn_wmma_f32_16x16x4_f32(false, a, false, b, (short)0, c,
                                                  false, false);
    }

    // C/D layout: VGPR r -> row = r + 8*half, col = idx (within tile)
    const float bv = bias[col];
    #pragma unroll
    for (int r = 0; r < 8; ++r) {
        const int row = row0 + r + 8 * half;
        float v = c[r] + bv;
        v = (v > 0.f) ? v : (__expf(v) - 1.f);   // ELU, alpha=1
        out[(size_t)row * D + col] = v;
    }
}

extern "C" void kernel_launch(void* const* d_in, const int* in_sizes, int n_in,
                              void* d_out, int out_size, void* d_ws, size_t ws_size,
                              hipStream_t stream)
{
    const float* feats = (const float*)d_in[0];
    const float* att   = (const float*)d_in[1];
    const float* W     = (const float*)d_in[2];
    const float* bias  = (const float*)d_in[3];
    const int*   seg   = (const int*)d_in[4];
    float* out = (float*)d_out;

    const int n_edges  = in_sizes[1];          // 640000
    const int n_nodes  = out_size / D;         // 40000
    const int n_blocks = (n_nodes + TILE_M - 1) / TILE_M;   // 2500

    grn_fused_kernel<<<n_blocks, THREADS, 0, stream>>>(feats, att, W, bias, seg,
                                                       out, n_edges);
}